// DCP_73332271612168
// MI455X (gfx1250) — compile-verified
//
#include <hip/hip_runtime.h>

typedef __attribute__((ext_vector_type(2))) float v2f;
typedef __attribute__((ext_vector_type(4))) float v4f;
typedef __attribute__((ext_vector_type(8))) float v8f;

// ---------------------------------------------------------------------------
// Wave-level fp32 reduction using the CDNA5 matrix pipe.
// A (16x4 f32): lanes 0-15 hold {K0,K1} = {p[lane], 0}; lanes 16-31 hold
// {K2,K3} = {p[lane], 0}.  With B = ones(4x16), D[m,n] = p[m] + p[m+16].
// Each lane sums its 8 D VGPRs: lanes 0-15 get Sum_{m=0..7}, lanes 16-31 get
// Sum_{m=8..15}.  One shfl_down(16) then puts the full wave sum in lane 0.
// EXEC must be all 1s at the call site (callers invoke this convergently).
// ---------------------------------------------------------------------------
__device__ __forceinline__ float wave_reduce_wmma(float v) {
    v2f a; a.x = v;    a.y = 0.0f;
    v2f b; b.x = 1.0f; b.y = 1.0f;
    v8f c = {};
    v8f d = __builtin_amdgcn_wmma_f32_16x16x4_f32(
        /*neg_a=*/false, a, /*neg_b=*/false, b,
        /*c_mod=*/(short)0, c, /*reuse_a=*/false, /*reuse_b=*/false);
    float s = (d[0] + d[1]) + (d[2] + d[3]) + (d[4] + d[5]) + (d[6] + d[7]);
    s += __shfl_down(s, 16, 32);   // lane0: full 32-lane sum
    return s;
}

// Block reduction: WMMA wave reduce, then LDS across (<=8) waves.
// Returns the block total in thread 0 (garbage elsewhere).
__device__ __forceinline__ float block_reduce(float v, float* lds) {
    float s = wave_reduce_wmma(v);
    const int wave = threadIdx.x >> 5;
    const int lane = threadIdx.x & 31;
    if (lane == 0) lds[wave] = s;
    __syncthreads();
    float t = 0.0f;
    if (wave == 0) {
        const int nwaves = blockDim.x >> 5;
        t = (lane < nwaves) ? lds[lane] : 0.0f;
        t += __shfl_down(t, 16, 32);
        t += __shfl_down(t, 8, 32);
        t += __shfl_down(t, 4, 32);
        t += __shfl_down(t, 2, 32);
        t += __shfl_down(t, 1, 32);
    }
    return t;
}

// ---------------------------------------------------------------------------
// Pass 1: streaming weighted reduction.
// x: [16, 3, 1024, 1024].  Each "group" = 4 consecutive columns (float4).
//   g = b*(1024*256) + row*256 + j4
// Per group: load 3 channel float4s (NT, single-use), componentwise min,
// abs, multiply by edge weights {2,3}, accumulate per-lane fp32.
// ---------------------------------------------------------------------------
__global__ __launch_bounds__(256)
void dcp_partial_kernel(const float* __restrict__ x,
                        float* __restrict__ part, int ngroups) {
    __shared__ float lds[8];
    constexpr int CH = 1024 * 1024;   // channel stride (elements)

    float acc = 0.0f;
    const int stride = gridDim.x * blockDim.x;
    for (int g = blockIdx.x * blockDim.x + threadIdx.x; g < ngroups; g += stride) {
        const int j4  = g & 255;          // column group (0..255)
        const int row = (g >> 8) & 1023;  // row (0..1023)
        const int b   = g >> 18;          // batch (0..15)

        const size_t base = (size_t)b * (3 * CH) + (size_t)row * 1024 + 4 * (size_t)j4;
        const v4f* p = (const v4f*)(x + base);

        const v4f c0 = __builtin_nontemporal_load(p);
        const v4f c1 = __builtin_nontemporal_load(p + (CH / 4));
        const v4f c2 = __builtin_nontemporal_load(p + (2 * CH / 4));

        const float m0 = fabsf(fminf(fminf(c0.x, c1.x), c2.x));
        const float m1 = fabsf(fminf(fminf(c0.y, c1.y), c2.y));
        const float m2 = fabsf(fminf(fminf(c0.z, c1.z), c2.z));
        const float m3 = fabsf(fminf(fminf(c0.w, c1.w), c2.w));

        // column coverage weights: col 0 / col 1023 -> 2, else 3
        const float w0 = (j4 == 0)   ? 2.0f : 3.0f;
        const float w3 = (j4 == 255) ? 2.0f : 3.0f;
        // row coverage weight
        const float wr = (row == 0 || row == 1023) ? 2.0f : 3.0f;

        float s = w0 * m0;
        s = fmaf(3.0f, m1, s);
        s = fmaf(3.0f, m2, s);
        s = fmaf(w3,   m3, s);
        acc = fmaf(wr, s, acc);
    }

    const float t = block_reduce(acc, lds);
    if (threadIdx.x == 0) part[blockIdx.x] = t;
}

// ---------------------------------------------------------------------------
// Pass 2: reduce per-block partials to the final scalar (deterministic).
// ---------------------------------------------------------------------------
__global__ __launch_bounds__(256)
void dcp_final_kernel(const float* __restrict__ part, int n,
                      float* __restrict__ out) {
    __shared__ float lds[8];
    float acc = 0.0f;
    for (int i = threadIdx.x; i < n; i += blockDim.x) acc += part[i];
    const float t = block_reduce(acc, lds);
    if (threadIdx.x == 0) out[0] = t;
}

extern "C" void kernel_launch(void* const* d_in, const int* in_sizes, int n_in,
                              void* d_out, int out_size, void* d_ws, size_t ws_size,
                              hipStream_t stream) {
    (void)in_sizes; (void)n_in; (void)out_size; (void)ws_size;
    const float* x = (const float*)d_in[0];   // [16,3,1024,1024] fp32
    float* part = (float*)d_ws;               // NBLK fp32 partials
    float* out  = (float*)d_out;              // scalar fp32

    constexpr int NGROUPS = 16 * 1024 * 256;  // pixels / 4
    constexpr int NBLK    = 2048;             // 16K waves in flight

    dcp_partial_kernel<<<NBLK, 256, 0, stream>>>(x, part, NGROUPS);
    dcp_final_kernel<<<1, 256, 0, stream>>>(part, NBLK, out);
}